// GCMCLayer_3959959847141
// MI455X (gfx1250) — compile-verified
//
#include <hip/hip_runtime.h>
#include <hip/hip_bf16.h>

typedef __attribute__((ext_vector_type(2))) float v2f;
typedef __attribute__((ext_vector_type(8))) float v8f;

#define N_DRUG    50000
#define N_DIS     50000
#define N_EDGES   500000
#define IN_UNITS  4096
#define MSG_SMALL 128
#define OUT_UNITS 256
#define BASIS_UNITS 4
#define N_RATINGS 2

// ---------------------------------------------------------------------------
// Kernel 0: zero the two 256-wide accumulators (contiguous region in ws).
// ---------------------------------------------------------------------------
__global__ __launch_bounds__(256) void k_zero(float* __restrict__ p, int n) {
    int stride = gridDim.x * blockDim.x;
    for (int i = blockIdx.x * blockDim.x + threadIdx.x; i < n; i += stride)
        p[i] = 0.0f;
}

// ---------------------------------------------------------------------------
// Kernel 1: W[r][i][m] = sum_b att[r][b] * basis[b][i][m]   (float4-vectorized)
// ---------------------------------------------------------------------------
__global__ __launch_bounds__(256) void k_basis(const float* __restrict__ att,
                                               const float* __restrict__ basis,
                                               float* __restrict__ W) {
    int idx = blockIdx.x * 256 + threadIdx.x;          // 262144 threads, 4 floats each
    int r   = idx / (IN_UNITS * 32);
    int rem = idx - r * (IN_UNITS * 32);
    int i   = rem >> 5;
    int m4  = (rem & 31) << 2;
    float4 acc = make_float4(0.f, 0.f, 0.f, 0.f);
#pragma unroll
    for (int b = 0; b < BASIS_UNITS; ++b) {
        float a = att[r * BASIS_UNITS + b];
        const float4 bs =
            *(const float4*)&basis[((size_t)(b * IN_UNITS + i)) * MSG_SMALL + m4];
        acc.x += a * bs.x; acc.y += a * bs.y; acc.z += a * bs.z; acc.w += a * bs.w;
    }
    *(float4*)&W[((size_t)(r * IN_UNITS + i)) * MSG_SMALL + m4] = acc;
}

// ---------------------------------------------------------------------------
// Kernel 2: G[rk][i][o] = sum_m W[r][i][m] * fc_w[o][k*128 + m]
//   M = 4096 (i), N = 256 (o), K = 128 (m), done with V_WMMA_F32_16X16X4_F32.
//   Block = 512 threads = 16 waves; each wave owns one 16-wide column tile
//   (16 waves * 16 = 256 = all of N). One block per 16-row tile per rk.
//   A tile (16x128 f32) staged in LDS, stride padded to 130 words so the 16
//   lanes of a half-wave hit distinct banks (130 % 64 == 2).
// ---------------------------------------------------------------------------
__global__ __launch_bounds__(512) void k_fold_fc(const float* __restrict__ W,
                                                 const float* __restrict__ fc_w,
                                                 float* __restrict__ G) {
    __shared__ float Ash[16 * 130];
    const int rk = blockIdx.y;            // 0..5 : r = rk/3, k = rk%3
    const int r  = rk / 3, k = rk % 3;
    const int row_base = blockIdx.x * 16;

    const float* A = W + (size_t)r * IN_UNITS * MSG_SMALL;
    for (int t = threadIdx.x; t < 16 * 128; t += 512) {
        int row = t >> 7, col = t & 127;
        Ash[row * 130 + col] = A[(size_t)(row_base + row) * MSG_SMALL + col];
    }
    __syncthreads();

    const int wave = threadIdx.x >> 5;    // wave32
    const int lane = threadIdx.x & 31;
    const int col_base = wave * 16;
    const int mrow = lane & 15;           // A row / B column selector (ISA layout)
    const int kh   = (lane >> 4) * 2;     // lanes 0-15: K=0,1 ; lanes 16-31: K=2,3
    // B[mk][n] = fc_w[n][k*128 + mk]; lane's column n = col_base + mrow
    const float* Bp = fc_w + (size_t)(col_base + mrow) * (3 * MSG_SMALL) + k * MSG_SMALL;

    v8f c = {};
#pragma unroll
    for (int kk = 0; kk < 32; ++kk) {     // 128 / 4 K-steps
        const int k0 = kk * 4 + kh;
        v2f a, b;
        a.x = Ash[mrow * 130 + k0];
        a.y = Ash[mrow * 130 + k0 + 1];
        const v2f bv = *(const v2f*)&Bp[k0];
        b.x = bv.x; b.y = bv.y;
        c = __builtin_amdgcn_wmma_f32_16x16x4_f32(
                /*neg_a=*/false, a, /*neg_b=*/false, b,
                /*c_mod=*/(short)0, c, /*reuse_a=*/false, /*reuse_b=*/false);
    }

    // C/D layout: VGPR i -> row M=i (lanes 0-15) / M=i+8 (lanes 16-31), N = lane&15
    float* D = G + ((size_t)rk * IN_UNITS + row_base) * OUT_UNITS + col_base;
    const int nn = lane & 15;
    const int mh = (lane >> 4) * 8;
#pragma unroll
    for (int i = 0; i < 8; ++i)
        D[(size_t)(mh + i) * OUT_UNITS + nn] = c[i];
}

// ---------------------------------------------------------------------------
// Kernel 3: edge scatter. One block (256 threads) per (edge, rating); thread o
// handles output channel o for both directions. Gathers 3 rows of the fused
// G table (L2-resident, 25 MB) and does fp32 atomic adds into the L2-resident
// accumulators.  acc_dis[d] += cj_drug[s] * sum_k G[r][k][drug_feat[s][k]]
//                acc_drug[s] += cj_dis[d] * sum_k G[r][k][dis_feat[d][k]]
// ---------------------------------------------------------------------------
__global__ __launch_bounds__(256) void k_scatter(const int*   __restrict__ edge_src,
                                                 const int*   __restrict__ edge_dst,
                                                 const int*   __restrict__ drug_feat,
                                                 const int*   __restrict__ dis_feat,
                                                 const float* __restrict__ cj_drug,
                                                 const float* __restrict__ cj_dis,
                                                 const float* __restrict__ G,
                                                 float* __restrict__ acc_drug,
                                                 float* __restrict__ acc_dis) {
    const int e = blockIdx.x;
    const int r = blockIdx.y;
    const int o = threadIdx.x;

    const int s = edge_src[(size_t)r * N_EDGES + e];
    const int d = edge_dst[(size_t)r * N_EDGES + e];
    const float* Gr = G + (size_t)r * 3 * IN_UNITS * OUT_UNITS;
    const size_t plane = (size_t)IN_UNITS * OUT_UNITS;

    // forward: drug s -> disease d
    {
        const int f0 = drug_feat[s * 3 + 0];
        const int f1 = drug_feat[s * 3 + 1];
        const int f2 = drug_feat[s * 3 + 2];
        const float v = Gr[(size_t)f0 * OUT_UNITS + o]
                      + Gr[plane     + (size_t)f1 * OUT_UNITS + o]
                      + Gr[2 * plane + (size_t)f2 * OUT_UNITS + o];
        atomicAdd(&acc_dis[(size_t)d * OUT_UNITS + o], v * cj_drug[s]);
    }
    // reverse: disease d -> drug s
    {
        const int g0 = dis_feat[d * 3 + 0];
        const int g1 = dis_feat[d * 3 + 1];
        const int g2 = dis_feat[d * 3 + 2];
        const float u = Gr[(size_t)g0 * OUT_UNITS + o]
                      + Gr[plane     + (size_t)g1 * OUT_UNITS + o]
                      + Gr[2 * plane + (size_t)g2 * OUT_UNITS + o];
        atomicAdd(&acc_drug[(size_t)s * OUT_UNITS + o], u * cj_dis[d]);
    }
}

// ---------------------------------------------------------------------------
// Kernel 4: out[n][o] = ci[n] * acc[n][o] + fc_b[o]   (both sides via grid.y)
// ---------------------------------------------------------------------------
__global__ __launch_bounds__(256) void k_finalize(const float* __restrict__ acc_drug,
                                                  const float* __restrict__ acc_dis,
                                                  const float* __restrict__ ci_drug,
                                                  const float* __restrict__ ci_dis,
                                                  const float* __restrict__ fc_b,
                                                  float* __restrict__ out) {
    const int side = blockIdx.y;                 // 0 = drug, 1 = dis
    const int n = blockIdx.x;
    const int o = threadIdx.x;
    const float* acc = side ? acc_dis : acc_drug;
    const float  ci  = side ? ci_dis[n] : ci_drug[n];
    float* op = out + (size_t)side * N_DRUG * OUT_UNITS;
    op[(size_t)n * OUT_UNITS + o] = ci * acc[(size_t)n * OUT_UNITS + o] + fc_b[o];
}

// ---------------------------------------------------------------------------
extern "C" void kernel_launch(void* const* d_in, const int* in_sizes, int n_in,
                              void* d_out, int out_size, void* d_ws, size_t ws_size,
                              hipStream_t stream) {
    const int*   drug_feat = (const int*)  d_in[0];
    const int*   dis_feat  = (const int*)  d_in[1];
    const int*   edge_src  = (const int*)  d_in[2];
    const int*   edge_dst  = (const int*)  d_in[3];
    const float* cj_drug   = (const float*)d_in[4];
    const float* ci_drug   = (const float*)d_in[5];
    const float* cj_dis    = (const float*)d_in[6];
    const float* ci_dis    = (const float*)d_in[7];
    const float* att       = (const float*)d_in[8];
    const float* basis     = (const float*)d_in[9];
    const float* fc_w      = (const float*)d_in[10];
    const float* fc_b      = (const float*)d_in[11];
    float* out = (float*)d_out;

    // workspace layout (floats):
    float* W        = (float*)d_ws;                                  // 2*4096*128   = 1,048,576
    float* G        = W + (size_t)N_RATINGS * IN_UNITS * MSG_SMALL;  // 6*4096*256   = 6,291,456
    float* acc_drug = G + (size_t)N_RATINGS * 3 * IN_UNITS * OUT_UNITS; // 50000*256
    float* acc_dis  = acc_drug + (size_t)N_DRUG * OUT_UNITS;            // 50000*256

    // 0) zero accumulators (acc_drug and acc_dis are contiguous)
    const int acc_total = (N_DRUG + N_DIS) * OUT_UNITS;
    k_zero<<<4096, 256, 0, stream>>>(acc_drug, acc_total);

    // 1) W = att @ basis
    k_basis<<<(N_RATINGS * IN_UNITS * MSG_SMALL / 4) / 256, 256, 0, stream>>>(att, basis, W);

    // 2) fold fc into the weight table: G[r][k] = W[r] @ fc_w[:, k*128:(k+1)*128]^T
    k_fold_fc<<<dim3(IN_UNITS / 16, N_RATINGS * 3), 512, 0, stream>>>(W, fc_w, G);

    // 3) edge scatter with fp32 atomics (256-wide instead of 384-wide)
    k_scatter<<<dim3(N_EDGES, N_RATINGS), 256, 0, stream>>>(
        edge_src, edge_dst, drug_feat, dis_feat, cj_drug, cj_dis, G, acc_drug, acc_dis);

    // 4) out = ci * acc + b  (drug then dis, concatenated in d_out)
    k_finalize<<<dim3(N_DRUG, 2), 256, 0, stream>>>(
        acc_drug, acc_dis, ci_drug, ci_dis, fc_b, out);
}